// GNN_51556787421768
// MI455X (gfx1250) — compile-verified
//
#include <hip/hip_runtime.h>
#include <hip/hip_bf16.h>
#include <cstdint>

// ---------------------------------------------------------------------------
// GraphSAGE forward on MI455X (gfx1250, wave32).
//
// Roofline: edge aggregation (~1.6M edges x 512B/layer) dominates at 23.3TB/s;
// the 6 dense [100k,128]x[128,128] GEMMs are ~20 GFLOP and essentially free,
// so we keep them in exact f32 using V_WMMA_F32_16X16X4_F32 (2048 FLOP/instr).
//
// GEMM data path: the 16x128 node tiles (agg + x, 8KB each, contiguous) are
// staged into LDS once per block by the Tensor Data Mover (TENSOR_LOAD_TO_LDS
// + s_wait_tensorcnt), removing the 8x redundant A-fragment global loads that
// all 8 waves of a block would otherwise issue. B-fragments (weights) stay in
// global: they are per-wave-unique and L2-resident (64KB vs 192MB L2).
// ---------------------------------------------------------------------------

#define D_FEAT 128
#define NGRAPH 64

typedef __attribute__((ext_vector_type(2))) float v2f;
typedef __attribute__((ext_vector_type(8))) float v8f;
typedef __attribute__((ext_vector_type(4))) unsigned int u32x4;
typedef __attribute__((ext_vector_type(8))) int i32x8;
typedef __attribute__((ext_vector_type(4))) int i32x4;

// ---------------- TDM: contiguous 1-D copy global -> LDS -------------------
// D# per CDNA5 ISA ch.8: group0 {count, lds_addr, global_addr, type=2},
// group1 {data_size=4B, tensor_dim0=n_elems (OOB reads return 0),
//         tensor_dim1=1, tile_dim0=tile_elems, tile_dim1=0 (1-D), stride}.
// This toolchain exposes the 6-arg (clang-23/therock) builtin:
//   (u32x4 g0, i32x8 g1, i32x4 g2, i32x4 g3, i32x8 extra, i32 cpol)
__device__ __forceinline__ void tdm_load_1d(const float* gsrc, float* lds_dst,
                                            unsigned n_elems, unsigned tile_elems) {
    unsigned long long ga = (unsigned long long)(uintptr_t)gsrc;
    unsigned lo = (unsigned)(uintptr_t)lds_dst;   // LDS byte offset (addr[31:0])
    u32x4 g0;
    g0[0] = 1u;                                           // count=1, no gather
    g0[1] = lo;                                           // lds_addr
    g0[2] = (unsigned)(ga & 0xFFFFFFFFu);                 // global_addr[31:0]
    g0[3] = (unsigned)((ga >> 32) & 0x01FFFFFFu)          // global_addr[56:32]
          | (2u << 30);                                   // type = 2 ("image")
    i32x8 g1;
    g1[0] = (int)(2u << 16);                              // data_size = 4 bytes
    g1[1] = (int)((n_elems & 0xFFFFu) << 16);             // tensor_dim0[15:0]
    g1[2] = (int)((n_elems >> 16) & 0xFFFFu)              // tensor_dim0[31:16]
          | (int)(1u << 16);                              // tensor_dim1 = 1
    g1[3] = (int)((tile_elems & 0xFFFFu) << 16);          // tile_dim0
    g1[4] = 0;                                            // tile_dim1=0 -> 1-D
    g1[5] = (int)n_elems;                                 // tensor_dim0_stride[31:0]
    g1[6] = 0;
    g1[7] = 0;
    i32x4 z4 = {0, 0, 0, 0};
    i32x8 z8 = {0, 0, 0, 0, 0, 0, 0, 0};
    __builtin_amdgcn_tensor_load_to_lds(g0, g1, z4, z4, z8, 0);
}

// ---------------------------- utility: zero fill ---------------------------
__global__ void gnn_fill_zero(float* __restrict__ p, long long n) {
    long long i = (long long)blockIdx.x * blockDim.x + threadIdx.x;
    if (i < n) p[i] = 0.0f;
}

// ------------------- per-edge scatter-add (wave per edge) ------------------
__global__ void gnn_edge_scatter(const float* __restrict__ feat,
                                 const int* __restrict__ ei,   // [2,E]
                                 float* __restrict__ agg,      // [N,128]
                                 int n_edges) {
    long long t = (long long)blockIdx.x * blockDim.x + threadIdx.x;
    int e = (int)(t >> 5);
    if (e >= n_edges) return;
    int lane = (int)(t & 31);
    int s = ei[e];              // src row
    int d = ei[n_edges + e];    // dst row
    const float4 v = *((const float4*)(feat + (size_t)s * D_FEAT) + lane);
    float* a = agg + (size_t)d * D_FEAT + lane * 4;
    atomicAdd(a + 0, v.x);
    atomicAdd(a + 1, v.y);
    atomicAdd(a + 2, v.z);
    atomicAdd(a + 3, v.w);
}

// ------------------------------ degree count -------------------------------
__global__ void gnn_edge_degree(const int* __restrict__ ei,
                                float* __restrict__ deg, int n_edges) {
    long long t = (long long)blockIdx.x * blockDim.x + threadIdx.x;
    if (t < n_edges) atomicAdd(&deg[ei[n_edges + (int)t]], 1.0f);
}

// --------------- fused SAGE layer: relu(agg/deg*Wl + bl + x*Wr) ------------
// Block = 256 threads = 8 waves. Block b owns node rows [16b,16b+16);
// wave w owns output columns [16w,16w+16). Node tiles staged via TDM into
// LDS; K loop in steps of 4 with two V_WMMA_F32_16X16X4_F32 per step.
//
// A 16x4 f32 frag (ISA 7.12.2): lane<16 holds M=lane, K={k0,k0+1};
//                               lane>=16 holds M=lane-16, K={k0+2,k0+3}.
// B 4x16 frag mirrored; C/D: VGPR r, lane<16 -> M=r, lane>=16 -> M=r+8,
// N = lane&15.
__global__ void gnn_sage_gemm(const float* __restrict__ xin,  // [N,128] root
                              const float* __restrict__ agg,  // [N,128] summed
                              const float* __restrict__ deg,  // [N]
                              const float* __restrict__ Wl,   // [128,128] k-major
                              const float* __restrict__ bl,   // [128]
                              const float* __restrict__ Wr,   // [128,128]
                              float* __restrict__ out,        // [N,128]
                              int n_nodes, int do_relu) {
    __shared__ float sAgg[16 * D_FEAT];
    __shared__ float sX[16 * D_FEAT];
    __shared__ float sInv[16];

    const int tid  = threadIdx.x;
    const int lane = tid & 31;
    const int wave = tid >> 5;               // 0..7 -> column tile
    const int hi   = lane >> 4;              // lane-half selects K+2 / M+8
    const int lm   = lane & 15;
    const int n0   = wave * 16;
    const int base = blockIdx.x * 16;

    if (tid < 16) {
        int r = base + tid;
        sInv[tid] = (r < n_nodes) ? 1.0f / fmaxf(deg[r], 1.0f) : 0.0f;
    }

    if (wave == 0) {
        // valid elements in this tile; TDM zero-fills reads past tensor_dim0
        int remel = (n_nodes - base) * D_FEAT;
        unsigned td = (remel >= 16 * D_FEAT) ? (unsigned)(16 * D_FEAT)
                                             : (unsigned)(remel > 0 ? remel : 0);
        tdm_load_1d(agg + (size_t)base * D_FEAT, sAgg, td, 16 * D_FEAT);
        tdm_load_1d(xin + (size_t)base * D_FEAT, sX,  td, 16 * D_FEAT);
        __builtin_amdgcn_s_wait_tensorcnt(0);
    }
    __syncthreads();

    const float inv = sInv[lm];
    v8f c = {};
    #pragma unroll 4
    for (int k0 = 0; k0 < D_FEAT; k0 += 4) {
        const int k = k0 + hi * 2;
        v2f aA = *(const v2f*)&sAgg[lm * D_FEAT + k];   // ds_load_b64
        v2f aX = *(const v2f*)&sX[lm * D_FEAT + k];
        aA = aA * inv;
        v2f bL, bR;
        bL[0] = Wl[(size_t)k * D_FEAT + n0 + lm];
        bL[1] = Wl[(size_t)(k + 1) * D_FEAT + n0 + lm];
        bR[0] = Wr[(size_t)k * D_FEAT + n0 + lm];
        bR[1] = Wr[(size_t)(k + 1) * D_FEAT + n0 + lm];
        c = __builtin_amdgcn_wmma_f32_16x16x4_f32(false, aA, false, bL,
                                                  (short)0, c, false, false);
        c = __builtin_amdgcn_wmma_f32_16x16x4_f32(false, aX, false, bR,
                                                  (short)0, c, false, false);
    }

    const float bias = bl[n0 + lm];
    #pragma unroll
    for (int r = 0; r < 8; ++r) {
        const int m = base + r + hi * 8;
        if (m < n_nodes) {
            float v = c[r] + bias;
            if (do_relu) v = fmaxf(v, 0.0f);
            out[(size_t)m * D_FEAT + n0 + lm] = v;
        }
    }
}

// -------------------- per-graph mean pool (scatter phase) ------------------
__global__ void gnn_pool_scatter(const float* __restrict__ h,
                                 const int* __restrict__ batch,
                                 float* __restrict__ pooled,   // [G,128]
                                 int n_nodes) {
    long long t = (long long)blockIdx.x * blockDim.x + threadIdx.x;
    int nidx = (int)(t >> 5);
    if (nidx >= n_nodes) return;
    int lane = (int)(t & 31);
    int g = batch[nidx];
    const float4 v = *((const float4*)(h + (size_t)nidx * D_FEAT) + lane);
    float* p = pooled + (size_t)g * D_FEAT + lane * 4;
    atomicAdd(p + 0, v.x);
    atomicAdd(p + 1, v.y);
    atomicAdd(p + 2, v.z);
    atomicAdd(p + 3, v.w);
}

__global__ void gnn_pool_count(const int* __restrict__ batch,
                               float* __restrict__ cnt, int n_nodes) {
    long long t = (long long)blockIdx.x * blockDim.x + threadIdx.x;
    if (t < n_nodes) atomicAdd(&cnt[batch[(int)t]], 1.0f);
}

// ------------------- final linear: [G,128]@[128,2] + bias ------------------
__global__ void gnn_final_linear(const float* __restrict__ pooled,
                                 const float* __restrict__ cnt,
                                 const float* __restrict__ Wlin, // [128,2]
                                 const float* __restrict__ blin, // [2]
                                 float* __restrict__ out) {      // [G,2]
    int t = threadIdx.x;              // 0..127
    int g = t >> 1, o = t & 1;
    float inv = 1.0f / fmaxf(cnt[g], 1.0f);
    float s = 0.0f;
    #pragma unroll 8
    for (int k = 0; k < D_FEAT; ++k)
        s += pooled[(size_t)g * D_FEAT + k] * inv * Wlin[k * 2 + o];
    out[g * 2 + o] = s + blin[o];
}

// ---------------------------------------------------------------------------
extern "C" void kernel_launch(void* const* d_in, const int* in_sizes, int n_in,
                              void* d_out, int out_size, void* d_ws, size_t ws_size,
                              hipStream_t stream) {
    const float* x     = (const float*)d_in[0];
    const int*   ei    = (const int*)d_in[1];
    const int*   batch = (const int*)d_in[2];
    const float* Wl1 = (const float*)d_in[3];
    const float* bl1 = (const float*)d_in[4];
    const float* Wr1 = (const float*)d_in[5];
    const float* Wl2 = (const float*)d_in[6];
    const float* bl2 = (const float*)d_in[7];
    const float* Wr2 = (const float*)d_in[8];
    const float* Wl3 = (const float*)d_in[9];
    const float* bl3 = (const float*)d_in[10];
    const float* Wr3 = (const float*)d_in[11];
    const float* Wlin = (const float*)d_in[12];
    const float* blin = (const float*)d_in[13];
    float* out = (float*)d_out;

    const int n = in_sizes[0] / D_FEAT;   // nodes
    const int e = in_sizes[1] / 2;        // edges

    // workspace carve-up
    char* ws = (char*)d_ws;
    const size_t feat_bytes = (size_t)n * D_FEAT * sizeof(float);
    float* agg = (float*)(ws);
    float* hA  = (float*)(ws + feat_bytes);
    float* hB  = (float*)(ws + 2 * feat_bytes);
    float* deg = (float*)(ws + 3 * feat_bytes);
    size_t deg_bytes = (((size_t)n * sizeof(float)) + 255) & ~(size_t)255;
    float* pooled = (float*)(ws + 3 * feat_bytes + deg_bytes);
    float* cnt = pooled + (size_t)NGRAPH * D_FEAT;

    const long long nfeat = (long long)n * D_FEAT;
    auto nblk = [](long long work, int bs) { return (unsigned)((work + bs - 1) / bs); };
    const dim3 B256(256);
    const unsigned gemm_blocks = nblk(n, 16);

    // degree once (graph identical across layers)
    gnn_fill_zero<<<nblk(n, 256), B256, 0, stream>>>(deg, n);
    gnn_edge_degree<<<nblk(e, 256), B256, 0, stream>>>(ei, deg, e);

    // layer 1: x -> hA (relu)
    gnn_fill_zero<<<nblk(nfeat, 256), B256, 0, stream>>>(agg, nfeat);
    gnn_edge_scatter<<<nblk((long long)e * 32, 256), B256, 0, stream>>>(x, ei, agg, e);
    gnn_sage_gemm<<<gemm_blocks, B256, 0, stream>>>(x, agg, deg, Wl1, bl1, Wr1, hA, n, 1);

    // layer 2: hA -> hB (relu)
    gnn_fill_zero<<<nblk(nfeat, 256), B256, 0, stream>>>(agg, nfeat);
    gnn_edge_scatter<<<nblk((long long)e * 32, 256), B256, 0, stream>>>(hA, ei, agg, e);
    gnn_sage_gemm<<<gemm_blocks, B256, 0, stream>>>(hA, agg, deg, Wl2, bl2, Wr2, hB, n, 1);

    // layer 3: hB -> hA (no relu)
    gnn_fill_zero<<<nblk(nfeat, 256), B256, 0, stream>>>(agg, nfeat);
    gnn_edge_scatter<<<nblk((long long)e * 32, 256), B256, 0, stream>>>(hB, ei, agg, e);
    gnn_sage_gemm<<<gemm_blocks, B256, 0, stream>>>(hB, agg, deg, Wl3, bl3, Wr3, hA, n, 0);

    // global mean pool + final linear
    gnn_fill_zero<<<nblk(NGRAPH * D_FEAT, 256), B256, 0, stream>>>(pooled, NGRAPH * D_FEAT);
    gnn_fill_zero<<<1, B256, 0, stream>>>(cnt, NGRAPH);
    gnn_pool_scatter<<<nblk((long long)n * 32, 256), B256, 0, stream>>>(hA, batch, pooled, n);
    gnn_pool_count<<<nblk(n, 256), B256, 0, stream>>>(batch, cnt, n);
    gnn_final_linear<<<1, dim3(NGRAPH * 2), 0, stream>>>(pooled, cnt, Wlin, blin, out);
}